// TransformSize_5231270166858
// MI455X (gfx1250) — compile-verified
//
#include <hip/hip_runtime.h>

// ---------------------------------------------------------------------------
// Problem constants (from reference): B=8, N1=16384, N2=4096, mlp=[256,128,128]
// ---------------------------------------------------------------------------
#define BATCH 8
#define N1 16384
#define N2 4096
#define C0 256     // layer0 cin
#define C1 128     // layer0 cout / layer1 cin
#define C2 128     // layer1 cout
#define NT 32      // MLP kernel: output columns per block (2 x 16-wide WMMA tiles)
#define XPAD 2     // LDS row padding (floats) -> lane-to-lane bank stride 2/64, conflict-free
#define CH 2048    // kNN kernel: p2 points staged in LDS per chunk

typedef __attribute__((ext_vector_type(2))) float v2f;
typedef __attribute__((ext_vector_type(8))) float v8f;

// ===========================================================================
// Kernel A: fused conv_bn_relu x2 (per-batch GEMMs) using V_WMMA_F32_16X16X4_F32
//   g[b, :, n0:n0+NT] = relu(s1*(W1 @ relu(s0*(W0 @ X) + t0)) + t1)
// Block: 256 threads (8 waves). Wave w computes rows [16w, 16w+16) x NT cols.
// Single K loop with TWO accumulators (one per 16-col tile): each A fragment
// feeds two independent WMMAs (no back-to-back RAW chain on one accumulator).
// LDS tiles are stored TRANSPOSED ([col][k], padded) so every B fragment is a
// single contiguous ds_load_b64 into an adjacent VGPR pair (no repack movs).
// ===========================================================================
__global__ __launch_bounds__(256) void mlp_fused_kernel(
    const float* __restrict__ f2,
    const float* __restrict__ W0, const float* __restrict__ b0,
    const float* __restrict__ gamma0, const float* __restrict__ beta0,
    const float* __restrict__ mean0, const float* __restrict__ var0,
    const float* __restrict__ W1, const float* __restrict__ b1,
    const float* __restrict__ gamma1, const float* __restrict__ beta1,
    const float* __restrict__ mean1, const float* __restrict__ var1,
    float* __restrict__ g)
{
    __shared__ __align__(16) float Xs[NT][C0 + XPAD];  // 33.0 KB, transposed [col][k]
    __shared__ __align__(16) float Hs[NT][C1 + XPAD];  // 16.6 KB, transposed [col][k]
    __shared__ float s0[C1], t0[C1], s1[C1], t1[C1];

    const int tid = threadIdx.x;
    const int b   = blockIdx.y;
    const int n0  = blockIdx.x * NT;

    // Fold BN (inference) + conv bias into per-channel affine: y = acc*s + t
    if (tid < C1) {
        float s = gamma0[tid] * rsqrtf(var0[tid] + 1e-5f);
        s0[tid] = s;
        t0[tid] = (b0[tid] - mean0[tid]) * s + beta0[tid];
    } else {
        int c = tid - C1;
        float s = gamma1[c] * rsqrtf(var1[c] + 1e-5f);
        s1[c] = s;
        t1[c] = (b1[c] - mean1[c]) * s + beta1[c];
    }

    // Cooperative load of X tile, transposed into LDS.
    // Global reads coalesced (32 consecutive floats per wave); LDS writes
    // conflict-free (bank stride 2 across the 32 cols of a wave).
    {
        const float* fb = f2 + (size_t)b * C0 * N2;
        int col = tid & 31;
        int rb  = tid >> 5;              // 0..7
        #pragma unroll
        for (int it = 0; it < C0 / 8; ++it) {
            int r = rb + it * 8;
            Xs[col][r] = fb[(size_t)r * N2 + n0 + col];
        }
    }
    __syncthreads();

    const int wave = tid >> 5;
    const int lane = tid & 31;
    const int half = lane >> 4;   // K-half selector for A/B fragments
    const int l16  = lane & 15;   // M (A) or N (B/C/D) index

    // -------- layer 0: H = relu(affine(W0 @ X)), K = 256 --------
    {
        const float* w0row = W0 + (size_t)(wave * 16 + l16) * C0;
        const float* x0 = &Xs[l16][0];        // B-tile column (N = l16)
        const float* x1 = &Xs[16 + l16][0];   // B-tile column (N = 16+l16)
        v8f acc0 = {}, acc1 = {};
        for (int k4 = 0; k4 < C0 / 4; ++k4) {
            const int k = k4 * 4 + half * 2;
            v2f a   = *(const v2f*)(w0row + k);   // global_load_b64
            v2f bv0 = *(const v2f*)(x0 + k);      // ds_load_b64 (contiguous)
            v2f bv1 = *(const v2f*)(x1 + k);      // ds_load_b64 (contiguous)
            acc0 = __builtin_amdgcn_wmma_f32_16x16x4_f32(
                false, a, false, bv0, (short)0, acc0, false, false);
            acc1 = __builtin_amdgcn_wmma_f32_16x16x4_f32(
                false, a, false, bv1, (short)0, acc1, false, false);
        }
        #pragma unroll
        for (int i = 0; i < 8; ++i) {
            int row = wave * 16 + i + half * 8;   // C/D layout: M = i + 8*half
            float s = s0[row], t = t0[row];
            Hs[l16][row]      = fmaxf(acc0[i] * s + t, 0.0f);
            Hs[16 + l16][row] = fmaxf(acc1[i] * s + t, 0.0f);
        }
    }
    __syncthreads();

    // -------- layer 1: G = relu(affine(W1 @ H)), K = 128 --------
    {
        float* gb = g + (size_t)b * C2 * N2;
        const float* w1row = W1 + (size_t)(wave * 16 + l16) * C1;
        const float* h0 = &Hs[l16][0];
        const float* h1 = &Hs[16 + l16][0];
        v8f acc0 = {}, acc1 = {};
        for (int k4 = 0; k4 < C1 / 4; ++k4) {
            const int k = k4 * 4 + half * 2;
            v2f a   = *(const v2f*)(w1row + k);
            v2f bv0 = *(const v2f*)(h0 + k);
            v2f bv1 = *(const v2f*)(h1 + k);
            acc0 = __builtin_amdgcn_wmma_f32_16x16x4_f32(
                false, a, false, bv0, (short)0, acc0, false, false);
            acc1 = __builtin_amdgcn_wmma_f32_16x16x4_f32(
                false, a, false, bv1, (short)0, acc1, false, false);
        }
        #pragma unroll
        for (int i = 0; i < 8; ++i) {
            int row = wave * 16 + i + half * 8;
            float s = s1[row], t = t1[row];
            gb[(size_t)row * N2 + n0 + l16]      = fmaxf(acc0[i] * s + t, 0.0f);
            gb[(size_t)row * N2 + n0 + 16 + l16] = fmaxf(acc1[i] * s + t, 0.0f);
        }
    }
}

// ===========================================================================
// Kernel B: 3-NN + weighted interpolation.
// One query per thread; p2 staged in LDS as float4 (x,y,z,|p|^2) in 2048-point
// chunks -> ONE ds_load_b128 broadcast per candidate point in the hot loop.
// Distances use the same expansion trick as the reference so that neighbor
// selection and the interpolation weights match numerically.
// Output writes out[b, c, q] are coalesced over q (thread index).
// ===========================================================================
__global__ __launch_bounds__(256) void knn_interp_kernel(
    const float* __restrict__ p1, const float* __restrict__ p2,
    const float* __restrict__ g, float* __restrict__ out)
{
    __shared__ __align__(16) float4 pts[CH];   // 32 KB

    const int tid = threadIdx.x;
    const int b   = blockIdx.y;
    const int q   = blockIdx.x * 256 + tid;

    const float* p1q = p1 + ((size_t)b * N1 + q) * 3;
    const float x1 = p1q[0], y1 = p1q[1], z1 = p1q[2];
    const float s1q = x1 * x1 + y1 * y1 + z1 * z1;

    float d0 = 3.402823466e+38f, d1 = 3.402823466e+38f, d2 = 3.402823466e+38f;
    int   i0 = 0, i1 = 0, i2 = 0;

    const float* p2b = p2 + (size_t)b * N2 * 3;
    for (int chunk = 0; chunk < N2 / CH; ++chunk) {
        __syncthreads();
        for (int i = tid; i < CH; i += 256) {
            const float* p = p2b + (size_t)(chunk * CH + i) * 3;
            float x = p[0], y = p[1], z = p[2];
            pts[i] = make_float4(x, y, z, x * x + y * y + z * z);
        }
        __syncthreads();
        #pragma unroll 4
        for (int i = 0; i < CH; ++i) {
            float4 p  = pts[i];
            float dot = x1 * p.x + y1 * p.y + z1 * p.z;
            float d   = s1q + p.w - 2.0f * dot;   // matches reference expansion
            int   j   = chunk * CH + i;
            if (d < d2) {
                if (d < d1) {
                    if (d < d0) { d2 = d1; i2 = i1; d1 = d0; i1 = i0; d0 = d; i0 = j; }
                    else        { d2 = d1; i2 = i1; d1 = d;  i1 = j; }
                } else          { d2 = d;  i2 = j; }
            }
        }
    }

    float r0 = 1.0f / (d0 + 1e-8f);
    float r1 = 1.0f / (d1 + 1e-8f);
    float r2 = 1.0f / (d2 + 1e-8f);
    float rs = r0 + r1 + r2;
    float w0 = r0 / rs, w1 = r1 / rs, w2 = r2 / rs;

    const float* gb = g + (size_t)b * C2 * N2;
    float* ob = out + (size_t)b * C2 * N1 + q;
    #pragma unroll 4
    for (int c = 0; c < C2; ++c) {
        const float* row = gb + (size_t)c * N2;     // L2-resident gathers
        ob[(size_t)c * N1] = w0 * row[i0] + w1 * row[i1] + w2 * row[i2];
    }
}

// ===========================================================================
extern "C" void kernel_launch(void* const* d_in, const int* in_sizes, int n_in,
                              void* d_out, int out_size, void* d_ws, size_t ws_size,
                              hipStream_t stream) {
    const float* p1     = (const float*)d_in[0];
    const float* p2     = (const float*)d_in[1];
    const float* f2     = (const float*)d_in[2];
    const float* W0     = (const float*)d_in[3];
    const float* b0     = (const float*)d_in[4];
    const float* gamma0 = (const float*)d_in[5];
    const float* beta0  = (const float*)d_in[6];
    const float* mean0  = (const float*)d_in[7];
    const float* var0   = (const float*)d_in[8];
    const float* W1     = (const float*)d_in[9];
    const float* b1     = (const float*)d_in[10];
    const float* gamma1 = (const float*)d_in[11];
    const float* beta1  = (const float*)d_in[12];
    const float* mean1  = (const float*)d_in[13];
    const float* var1   = (const float*)d_in[14];

    float* g = (float*)d_ws;   // [B, 128, 4096] intermediate = 16.78 MB (L2-resident)

    dim3 gridA(N2 / NT, BATCH);   // 128 x 8 blocks
    mlp_fused_kernel<<<gridA, 256, 0, stream>>>(
        f2, W0, b0, gamma0, beta0, mean0, var0,
        W1, b1, gamma1, beta1, mean1, var1, g);

    dim3 gridB(N1 / 256, BATCH);  // 64 x 8 blocks
    knn_interp_kernel<<<gridB, 256, 0, stream>>>(p1, p2, g, (float*)d_out);
}